// TrainableGATLayer_18700287607517
// MI455X (gfx1250) — compile-verified
//
#include <hip/hip_runtime.h>
#include <math.h>

typedef float v2f __attribute__((ext_vector_type(2)));
typedef float v8f __attribute__((ext_vector_type(8)));

#define BDIM     4
#define NNODES   20000
#define FDIM     128
#define EDGES    320000
#define HDIM     128
#define ODIM     128
#define MTOT     (BDIM * NNODES)      /* 80000 flattened nodes        */
#define ETOT     (EDGES + MTOT)       /* 400000 unique edge work items */
#define NEG_SLOPE 0.2f

/* ---- monotonic uint encoding of float for atomicMax segment-max ---- */
__device__ __forceinline__ unsigned enc_f32(float f) {
  unsigned u = __float_as_uint(f);
  return (u & 0x80000000u) ? ~u : (u | 0x80000000u);
}
__device__ __forceinline__ float dec_f32(unsigned k) {
  unsigned u = (k & 0x80000000u) ? (k ^ 0x80000000u) : ~k;
  return __uint_as_float(u);
}

/* ================= init: zero agg (lives in d_out), denom, -inf keys ===== */
__global__ __launch_bounds__(256)
void gat_init(float* __restrict__ agg, float* __restrict__ denom,
              unsigned* __restrict__ m_keys) {
  const size_t i = (size_t)blockIdx.x * blockDim.x + threadIdx.x;
  if (i < (size_t)MTOT * HDIM) agg[i] = 0.0f;
  if (i < (size_t)MTOT) {
    denom[i]  = 0.0f;
    m_keys[i] = 0x007FFFFFu;   /* enc(-inf) */
  }
}

/* ================= xl = x@Wl^T, xr = x@Wr^T via V_WMMA_F32_16X16X4_F32 ==== */
/* one wave per 16-row tile; A staged in LDS; 8 N-tiles x 32 K-steps x 2 mats */
__global__ __launch_bounds__(32)
void gat_gemm_xlxr(const float* __restrict__ x, const float* __restrict__ Wl,
                   const float* __restrict__ Wr, float* __restrict__ xl,
                   float* __restrict__ xr) {
  __shared__ float tileA[16 * FDIM];                 /* 8 KB */
  const int row0 = blockIdx.x * 16;
  const int lane = threadIdx.x;

  /* coalesced stage of 16x128 f32 tile */
  const float4* s4 = (const float4*)(x + (size_t)row0 * FDIM);
  float4* l4 = (float4*)tileA;
#pragma unroll
  for (int i = 0; i < 16; ++i) l4[lane + 32 * i] = s4[lane + 32 * i];
  __syncthreads();

  const int mrow  = lane & 15;
  const int khalf = (lane >> 4) * 2;     /* lanes 0-15: K0/K1, 16-31: K2/K3 */
  const int nlo   = lane & 15;
  const int mhi   = (lane >> 4) * 8;

  for (int mat = 0; mat < 2; ++mat) {
    const float* W  = mat ? Wr : Wl;
    float* out      = mat ? xr : xl;
    v8f acc[8];
#pragma unroll
    for (int n = 0; n < 8; ++n) acc[n] = (v8f){0,0,0,0,0,0,0,0};

    for (int k0 = 0; k0 < FDIM; k0 += 4) {
      v2f a;
      a.x = tileA[mrow * FDIM + k0 + khalf];
      a.y = tileA[mrow * FDIM + k0 + khalf + 1];
#pragma unroll
      for (int n = 0; n < 8; ++n) {
        const float* wrow = W + (size_t)(n * 16 + mrow) * FDIM + k0 + khalf;
        v2f b; b.x = wrow[0]; b.y = wrow[1];      /* B[k][n] = W[n][k]      */
        acc[n] = __builtin_amdgcn_wmma_f32_16x16x4_f32(
            false, a, false, b, (short)0, acc[n], false, false);
      }
    }
#pragma unroll
    for (int n = 0; n < 8; ++n)
#pragma unroll
      for (int r = 0; r < 8; ++r)
        out[(size_t)(row0 + mhi + r) * HDIM + n * 16 + nlo] = acc[n][r];
  }
}

/* ================= edge pass 1: logits + segment max =================== */
/* one edge per wave; lane l owns float4 slice [4l..4l+3] of H=128         */
__global__ __launch_bounds__(256)
void gat_edge_logits(const int* __restrict__ ei, const float* __restrict__ xl,
                     const float* __restrict__ xr, const float* __restrict__ att,
                     float* __restrict__ e_buf, unsigned* __restrict__ m_keys) {
  const int wave = blockIdx.x * (blockDim.x >> 5) + (threadIdx.x >> 5);
  const int lane = threadIdx.x & 31;
  if (wave >= ETOT) return;
  int s, d;
  if (wave < EDGES) { s = ei[wave]; d = ei[EDGES + wave]; }
  else              { s = wave - EDGES; d = s; }

  const float4 a  = ((const float4*)(xl + (size_t)s * HDIM))[lane];
  const float4 b  = ((const float4*)(xr + (size_t)d * HDIM))[lane];
  const float4 at = ((const float4*)att)[lane];
  float4 v = {a.x + b.x, a.y + b.y, a.z + b.z, a.w + b.w};
  v.x = v.x > 0.f ? v.x : NEG_SLOPE * v.x;
  v.y = v.y > 0.f ? v.y : NEG_SLOPE * v.y;
  v.z = v.z > 0.f ? v.z : NEG_SLOPE * v.z;
  v.w = v.w > 0.f ? v.w : NEG_SLOPE * v.w;
  float p = v.x * at.x + v.y * at.y + v.z * at.z + v.w * at.w;
#pragma unroll
  for (int off = 16; off > 0; off >>= 1) p += __shfl_xor(p, off, 32);
  if (lane == 0) {
    e_buf[wave] = p;
    atomicMax(&m_keys[d], enc_f32(p));
  }
}

/* ================= edge pass 2: exp + weighted denom =================== */
__global__ __launch_bounds__(256)
void gat_edge_denom(const int* __restrict__ ei, float* __restrict__ e_buf,
                    const unsigned* __restrict__ m_keys, float* __restrict__ denom) {
  const int t = blockIdx.x * blockDim.x + threadIdx.x;
  if (t >= ETOT) return;
  const int d   = (t < EDGES) ? ei[EDGES + t] : (t - EDGES);
  const float w = (t < EDGES) ? (float)BDIM : 1.0f;   /* duplicate multiplicity */
  const float ex = __expf(e_buf[t] - dec_f32(m_keys[d]));
  e_buf[t] = ex;                                      /* reuse buffer in place  */
  unsafeAtomicAdd(&denom[d], w * ex);
}

/* ================= edge pass 3: weighted scatter-add =================== */
__global__ __launch_bounds__(256)
void gat_edge_scatter(const int* __restrict__ ei, const float* __restrict__ xl,
                      const float* __restrict__ e_buf, const float* __restrict__ denom,
                      float* __restrict__ agg) {
  const int wave = blockIdx.x * (blockDim.x >> 5) + (threadIdx.x >> 5);
  const int lane = threadIdx.x & 31;
  if (wave >= ETOT) return;
  int s, d; float w;
  if (wave < EDGES) { s = ei[wave]; d = ei[EDGES + wave]; w = (float)BDIM; }
  else              { s = wave - EDGES; d = s; w = 1.0f; }

  const float coeff = w * e_buf[wave] / denom[d];
  const float4 v = ((const float4*)(xl + (size_t)s * HDIM))[lane];
  float* base = agg + (size_t)d * HDIM + lane * 4;
  unsafeAtomicAdd(base + 0, coeff * v.x);
  unsafeAtomicAdd(base + 1, coeff * v.y);
  unsafeAtomicAdd(base + 2, coeff * v.z);
  unsafeAtomicAdd(base + 3, coeff * v.w);
}

/* ================= out = (agg + bias_gat) @ fc_w^T + fc_b, in place ==== */
__global__ __launch_bounds__(32)
void gat_gemm_out(float* __restrict__ io, const float* __restrict__ fc_w,
                  const float* __restrict__ bias_gat, const float* __restrict__ fc_b) {
  __shared__ float tileA[16 * HDIM];
  const int row0 = blockIdx.x * 16;
  const int lane = threadIdx.x;

  const float4* s4 = (const float4*)(io + (size_t)row0 * HDIM);
  const float4* b4 = (const float4*)bias_gat;
  float4* l4 = (float4*)tileA;
#pragma unroll
  for (int i = 0; i < 16; ++i) {
    const int j = lane + 32 * i;
    float4 t = s4[j], bb = b4[j & 31];
    t.x += bb.x; t.y += bb.y; t.z += bb.z; t.w += bb.w;
    l4[j] = t;
  }
  __syncthreads();      /* full A tile read before in-place overwrite below */

  const int mrow  = lane & 15;
  const int khalf = (lane >> 4) * 2;
  const int nlo   = lane & 15;
  const int mhi   = (lane >> 4) * 8;

  v8f acc[8];
#pragma unroll
  for (int n = 0; n < 8; ++n) acc[n] = (v8f){0,0,0,0,0,0,0,0};
  for (int k0 = 0; k0 < HDIM; k0 += 4) {
    v2f a;
    a.x = tileA[mrow * HDIM + k0 + khalf];
    a.y = tileA[mrow * HDIM + k0 + khalf + 1];
#pragma unroll
    for (int n = 0; n < 8; ++n) {
      const float* wrow = fc_w + (size_t)(n * 16 + mrow) * HDIM + k0 + khalf;
      v2f b; b.x = wrow[0]; b.y = wrow[1];
      acc[n] = __builtin_amdgcn_wmma_f32_16x16x4_f32(
          false, a, false, b, (short)0, acc[n], false, false);
    }
  }
#pragma unroll
  for (int n = 0; n < 8; ++n) {
    const float fb = fc_b[n * 16 + nlo];
#pragma unroll
    for (int r = 0; r < 8; ++r)
      io[(size_t)(row0 + mhi + r) * ODIM + n * 16 + nlo] = acc[n][r] + fb;
  }
}

/* ======================================================================= */
extern "C" void kernel_launch(void* const* d_in, const int* in_sizes, int n_in,
                              void* d_out, int out_size, void* d_ws, size_t ws_size,
                              hipStream_t stream) {
  (void)in_sizes; (void)n_in; (void)out_size; (void)ws_size;
  const float* x        = (const float*)d_in[0];
  const int*   ei       = (const int*)  d_in[1];   /* [2,E]: src then dst   */
  const float* Wl       = (const float*)d_in[2];
  const float* Wr       = (const float*)d_in[3];
  const float* att      = (const float*)d_in[4];
  const float* bias_gat = (const float*)d_in[5];
  const float* fc_w     = (const float*)d_in[6];
  const float* fc_b     = (const float*)d_in[7];

  /* workspace layout (floats): xl | xr | e_buf | denom | m_keys */
  float* ws     = (float*)d_ws;
  float* xl     = ws;
  float* xr     = xl + (size_t)MTOT * HDIM;
  float* e_buf  = xr + (size_t)MTOT * HDIM;
  float* denom  = e_buf + (size_t)ETOT;
  unsigned* m_keys = (unsigned*)(denom + (size_t)MTOT);
  float* agg    = (float*)d_out;                 /* agg lives in d_out      */

  const int initN = ((size_t)MTOT * HDIM + 255) / 256;
  gat_init<<<initN, 256, 0, stream>>>(agg, denom, m_keys);

  gat_gemm_xlxr<<<MTOT / 16, 32, 0, stream>>>(x, Wl, Wr, xl, xr);

  gat_edge_logits<<<ETOT / 8, 256, 0, stream>>>(ei, xl, xr, att, e_buf, m_keys);

  gat_edge_denom<<<(ETOT + 255) / 256, 256, 0, stream>>>(ei, e_buf, m_keys, denom);

  gat_edge_scatter<<<ETOT / 8, 256, 0, stream>>>(ei, xl, e_buf, denom, agg);

  gat_gemm_out<<<MTOT / 16, 32, 0, stream>>>(agg, fc_w, bias_gat, fc_b);
}